// MTRL_DGN_71167608094873
// MI455X (gfx1250) — compile-verified
//
#include <hip/hip_runtime.h>
#include <hip/hip_bf16.h>

// Collapsed computation (see analysis): per batch row b,
//   v0  = tanh(x_b @ Wv0 + bv0)
//   v1  = tanh(v0  @ Wv1 + bv1)
//   c   = [x_b | v0 | v1]                      (384)
//   a0  = relu(c  @ W0 + b0)                   (128)
//   a1  = relu(a0 @ W1 + b1)                   (128)
//   out = a1 @ W2 + b2                         (64)
// This equals reference(...)[:,0,:] exactly (softmax rows sum to 1, v rows are
// identical across tasks, so attention is the identity on v).

typedef float wm_v2f __attribute__((ext_vector_type(2)));
typedef float wm_v8f __attribute__((ext_vector_type(8)));

#define B_BATCH 256
#define D_IN    128
#define H_DIM   128
#define O_DIM   64
#define C_DIM   384
#define PITCH_C 386   // 384 + 2 pad (pitch%64==2 -> conflict-free A reads)
#define PITCH_A 130   // 128 + 2 pad

// One 16x16 WMMA-accumulated tile: D = A(16xK from LDS) * B(Kx16 from global).
// f32 WMMA 16x16x4 fragment layouts (CDNA5 ISA 7.12.2):
//   A: lane L, vgpr v -> A[L%16][kb + v + 2*(L/16)]
//   B: lane L, vgpr v -> B[kb + v + 2*(L/16)][n0 + L%16]
//   C: lane L, vgpr v -> C[v + 8*(L/16)][n0 + L%16]
__device__ __forceinline__ wm_v8f tile_gemm(const float* Abase, int apitch,
                                            const float* __restrict__ Bmat, int ldb,
                                            int n0, int K, int lane)
{
    const int lhalf = lane >> 4;
    const int lmod  = lane & 15;
    wm_v8f acc = {0.f, 0.f, 0.f, 0.f, 0.f, 0.f, 0.f, 0.f};
    const float* arow = Abase + lmod * apitch + 2 * lhalf;            // LDS
    const float* bcol = Bmat + (size_t)(2 * lhalf) * ldb + n0 + lmod; // global
    #pragma unroll 4
    for (int kb = 0; kb < K; kb += 4) {
        wm_v2f a;
        a.x = arow[kb];
        a.y = arow[kb + 1];
        wm_v2f b;
        b.x = bcol[(size_t)kb * ldb];
        b.y = bcol[(size_t)(kb + 1) * ldb];
        // 8 args: (neg_a, A, neg_b, B, c_mod, C, reuse_a, reuse_b)
        acc = __builtin_amdgcn_wmma_f32_16x16x4_f32(false, a, false, b,
                                                    (short)0, acc, false, false);
    }
    return acc;
}

// Apply bias (+ optional activation) and scatter the C tile into LDS row-major.
__device__ __forceinline__ void tile_store_lds(float* dst, int pitch, int n0,
                                               wm_v8f acc, const float* __restrict__ bias,
                                               int lane, int act_mode)
{
    const int lhalf = lane >> 4;
    const int lmod  = lane & 15;
    const float bb  = bias[n0 + lmod];
    #pragma unroll
    for (int v = 0; v < 8; ++v) {
        float val = acc[v] + bb;
        if (act_mode == 1) val = tanhf(val);
        if (act_mode == 2) val = fmaxf(val, 0.0f);
        dst[(v + 8 * lhalf) * pitch + n0 + lmod] = val;
    }
}

__global__ __launch_bounds__(256, 1) void MTRL_DGN_71167608094873_kernel(
    const float* __restrict__ x,    // [256,128]
    const float* __restrict__ wv,   // [2,128,128]
    const float* __restrict__ bv,   // [2,128]
    const float* __restrict__ w0,   // [384,128]
    const float* __restrict__ b0v,  // [128]
    const float* __restrict__ w1,   // [128,128]
    const float* __restrict__ b1v,  // [128]
    const float* __restrict__ w2,   // [128,64]
    const float* __restrict__ b2v,  // [64]
    float* __restrict__ out)        // [256,64]
{
    __shared__ float cbuf[16 * PITCH_C]; // [ x | v0 | v1 ] for 16 batch rows
    __shared__ float act0[16 * PITCH_A];
    __shared__ float act1[16 * PITCH_A];

    const int tid  = threadIdx.x;
    const int wave = tid >> 5;          // 8 waves: one 16-wide N tile each
    const int lane = tid & 31;
    const int brow = blockIdx.x * 16;   // 16 batch rows per workgroup
    const int n0   = wave * 16;

    // Stage 0: coalesced load of the X tile into cbuf[:, 0:128]
    for (int i = tid; i < 16 * D_IN; i += 256) {
        const int m = i >> 7;
        const int k = i & (D_IN - 1);
        cbuf[m * PITCH_C + k] = x[(brow + m) * D_IN + k];
    }
    __syncthreads();

    // Stage 1: v0 = tanh(X @ Wv0 + bv0)  -> cbuf[:, 128:256]
    {
        wm_v8f acc = tile_gemm(cbuf, PITCH_C, wv, H_DIM, n0, D_IN, lane);
        tile_store_lds(cbuf + H_DIM, PITCH_C, n0, acc, bv, lane, /*tanh*/1);
    }
    __syncthreads();

    // Stage 2: v1 = tanh(v0 @ Wv1 + bv1) -> cbuf[:, 256:384]
    {
        wm_v8f acc = tile_gemm(cbuf + H_DIM, PITCH_C,
                               wv + D_IN * H_DIM, H_DIM, n0, H_DIM, lane);
        tile_store_lds(cbuf + 2 * H_DIM, PITCH_C, n0, acc, bv + H_DIM, lane, 1);
    }
    __syncthreads();

    // Stage 3: a0 = relu(c @ W0 + b0), K = 384
    {
        wm_v8f acc = tile_gemm(cbuf, PITCH_C, w0, H_DIM, n0, C_DIM, lane);
        tile_store_lds(act0, PITCH_A, n0, acc, b0v, lane, /*relu*/2);
    }
    __syncthreads();

    // Stage 4: a1 = relu(a0 @ W1 + b1)
    {
        wm_v8f acc = tile_gemm(act0, PITCH_A, w1, H_DIM, n0, H_DIM, lane);
        tile_store_lds(act1, PITCH_A, n0, acc, b1v, lane, 2);
    }
    __syncthreads();

    // Stage 5: out = a1 @ W2 + b2  (N=64 -> waves 0..3; wave-uniform branch
    // keeps EXEC all-ones inside the WMMA region)
    if (wave < 4) {
        wm_v8f acc = tile_gemm(act1, PITCH_A, w2, O_DIM, n0, H_DIM, lane);
        const int lhalf = lane >> 4;
        const int lmod  = lane & 15;
        const float bb  = b2v[n0 + lmod];
        #pragma unroll
        for (int v = 0; v < 8; ++v) {
            const int m = v + 8 * lhalf;
            out[(brow + m) * O_DIM + n0 + lmod] = acc[v] + bb;
        }
    }
}

extern "C" void kernel_launch(void* const* d_in, const int* in_sizes, int n_in,
                              void* d_out, int out_size, void* d_ws, size_t ws_size,
                              hipStream_t stream) {
    // setup_inputs order:
    // 0:x 1:mask 2:wq 3:bq 4:wk 5:bk 6:wv 7:bv 8:w0 9:b0 10:w1 11:b1 12:w2 13:b2
    const float* x   = (const float*)d_in[0];
    const float* wv  = (const float*)d_in[6];
    const float* bv  = (const float*)d_in[7];
    const float* w0  = (const float*)d_in[8];
    const float* b0v = (const float*)d_in[9];
    const float* w1  = (const float*)d_in[10];
    const float* b1v = (const float*)d_in[11];
    const float* w2  = (const float*)d_in[12];
    const float* b2v = (const float*)d_in[13];
    float* out = (float*)d_out;

    dim3 grid(B_BATCH / 16);   // 16 workgroups
    dim3 block(256);           // 8 wave32
    MTRL_DGN_71167608094873_kernel<<<grid, block, 0, stream>>>(
        x, wv, bv, w0, b0v, w1, b1v, w2, b2v, out);
}